// Model_50895362457700
// MI455X (gfx1250) — compile-verified
//
#include <hip/hip_runtime.h>
#include <math.h>

#define BB 4
#define LL 256
#define HH 768
#define TT 16
#define NCH 8

typedef float v2f __attribute__((ext_vector_type(2)));
typedef float v8f __attribute__((ext_vector_type(8)));

// workspace offsets (in floats)
#define OFF_EMB1   0
#define OFF_APRIME (OFF_EMB1 + BB*LL*HH)        // emb1: 786432
#define OFF_BM     (OFF_APRIME + BB*LL*LL)      // A':   262144
#define OFF_WCATT  (OFF_BM + BB*LL*LL)          // Bm:   262144
#define OFF_TRIP   (OFF_WCATT + 16*LL)          // WcatT: 4096
// trip buffer: BB*TT*2*HH = 98304 floats. total ~5.7MB

// ---------------------------------------------------------------------------
// Kernel 1: LayerNorm over H=768 for each of B*L rows
// ---------------------------------------------------------------------------
__global__ void k_ln0(const float* __restrict__ x, const float* __restrict__ g,
                      const float* __restrict__ be, float* __restrict__ y) {
  int row = blockIdx.x;
  int tid = threadIdx.x;
  const float* xr = x + row * HH;
  float* yr = y + row * HH;
  float v0 = xr[tid], v1 = xr[tid + 256], v2 = xr[tid + 512];
  float s = v0 + v1 + v2;
  float s2 = v0 * v0 + v1 * v1 + v2 * v2;
  __shared__ float rs[256], rs2[256];
  rs[tid] = s; rs2[tid] = s2;
  __syncthreads();
  for (int off = 128; off > 0; off >>= 1) {
    if (tid < off) { rs[tid] += rs[tid + off]; rs2[tid] += rs2[tid + off]; }
    __syncthreads();
  }
  float mean = rs[0] * (1.0f / HH);
  float var = rs2[0] * (1.0f / HH) - mean * mean;
  float rstd = rsqrtf(var + 1e-5f);
  yr[tid]       = (v0 - mean) * rstd * g[tid]       + be[tid];
  yr[tid + 256] = (v1 - mean) * rstd * g[tid + 256] + be[tid + 256];
  yr[tid + 512] = (v2 - mean) * rstd * g[tid + 512] + be[tid + 512];
}

// ---------------------------------------------------------------------------
// Kernel 2: triplet enhance (tiny T=16 GAT), one block per batch
// ---------------------------------------------------------------------------
__global__ void k_triplet(float* __restrict__ emb,          // [B,L,H] (mutated)
                          const int* __restrict__ a_st, const int* __restrict__ a_ed,
                          const int* __restrict__ o_st, const int* __restrict__ o_ed,
                          const int* __restrict__ sid,
                          const float* __restrict__ E,      // [2,H]
                          const float* __restrict__ Wtp,    // [2H+3,H]
                          const float* __restrict__ btp,    // [H]
                          const float* __restrict__ Wa,     // [3H]
                          const float* __restrict__ ba,     // [1]
                          const float* __restrict__ Wg,     // [2H,H]
                          const float* __restrict__ bg,     // [H]
                          float* __restrict__ tripbuf)      // [B][T][2H]
{
  int b = blockIdx.x;
  int tid = threadIdx.x;
  float* reg = tripbuf + b * TT * 2 * HH;
  float* embb = emb + b * LL * HH;
  const int* ast = a_st + b * TT; const int* aed = a_ed + b * TT;
  const int* ost = o_st + b * TT; const int* oed = o_ed + b * TT;
  const int* sd  = sid  + b * TT;

  __shared__ float sNf[TT][HH];       // 48 KB
  __shared__ float sSim[TT][TT];
  __shared__ float sP0[TT], sP1[TT], sP2[2];
  __shared__ float sScore[TT][TT * 2];
  __shared__ float sAttn[TT][TT * 2];
  __shared__ int   sM[TT][TT * 2];
  __shared__ int   sHasIn[TT];
  __shared__ float sNorm[TT];

  // (a) span means -> reg[t][0:H]=asp, reg[t][H:2H]=opi
  for (int idx = tid; idx < TT * 2 * HH; idx += 256) {
    int t = idx / (2 * HH);
    int c = idx % (2 * HH);
    int st, ed, h;
    if (c < HH) { st = ast[t]; ed = aed[t]; h = c; }
    else        { st = ost[t]; ed = oed[t]; h = c - HH; }
    float s = 0.f;
    for (int l = st; l <= ed; ++l) s += embb[l * HH + h];
    reg[idx] = s / (float)(ed - st + 1);
  }
  __syncthreads();
  // (b) nf = [asp|opi|onehot] @ W_tp + b_tp
  for (int idx = tid; idx < TT * HH; idx += 256) {
    int t = idx / HH, h = idx % HH;
    const float* rt = reg + t * 2 * HH;
    float acc = btp[h] + Wtp[(2 * HH + (sd[t] - 2)) * HH + h];
    for (int c = 0; c < 2 * HH; ++c) acc += rt[c] * Wtp[c * HH + h];
    sNf[t][h] = acc;
  }
  __syncthreads();
  // (c) norms
  if (tid < TT) {
    float s = 0.f;
    for (int h = 0; h < HH; ++h) { float v = sNf[tid][h]; s += v * v; }
    sNorm[tid] = fmaxf(sqrtf(s), 1e-8f);
  }
  __syncthreads();
  // (d) cosine sim
  if (tid < TT * TT) {
    int t = tid / TT, u = tid % TT;
    float s = 0.f;
    for (int h = 0; h < HH; ++h) s += sNf[t][h] * sNf[u][h];
    sSim[t][u] = s / (sNorm[t] * sNorm[u]);
  }
  // (e) attention score components p0[t], p1[s], p2[e]
  if (tid < TT) {
    float s = 0.f;
    for (int h = 0; h < HH; ++h) { float v = sNf[tid][h]; v = v > 0.f ? v : 0.2f * v; s += v * Wa[h]; }
    sP0[tid] = s;
  } else if (tid < 2 * TT) {
    int t = tid - TT; float s = 0.f;
    for (int h = 0; h < HH; ++h) { float v = sNf[t][h]; v = v > 0.f ? v : 0.2f * v; s += v * Wa[HH + h]; }
    sP1[t] = s;
  } else if (tid < 2 * TT + 2) {
    int e = tid - 2 * TT; float s = 0.f;
    for (int h = 0; h < HH; ++h) { float v = E[e * HH + h]; v = v > 0.f ? v : 0.2f * v; s += v * Wa[2 * HH + h]; }
    sP2[e] = s;
  }
  __syncthreads();
  // (f) edge masks + masked scores: emask[t][s][e] = m_e[s][t]
  if (tid < TT * TT * 2) {
    int t = tid / (TT * 2);
    int r = tid % (TT * 2);
    int s2 = r >> 1, e = r & 1;
    bool ok = (sSim[s2][t] > 0.f) && (s2 != t);
    bool same = (e == 0) ? ((ast[s2] == ast[t]) && (aed[s2] == aed[t]))
                         : ((ost[s2] == ost[t]) && (oed[s2] == oed[t]));
    int m = (ok && same) ? 1 : 0;
    sM[t][r] = m;
    sScore[t][r] = m ? (sP0[t] + sP1[s2] + sP2[e] + ba[0]) : -1e9f;
  }
  __syncthreads();
  // (g) softmax over 32 (s,e) per target t
  if (tid < TT) {
    float mx = -1e30f; int any = 0;
    for (int r = 0; r < 32; ++r) { mx = fmaxf(mx, sScore[tid][r]); any |= sM[tid][r]; }
    float sm = 0.f;
    for (int r = 0; r < 32; ++r) { float ex = expf(sScore[tid][r] - mx); sAttn[tid][r] = ex; sm += ex; }
    float inv = 1.f / sm;
    for (int r = 0; r < 32; ++r) sAttn[tid][r] *= inv;
    sHasIn[tid] = any;
  }
  __syncthreads();
  // (h) agg[t][0:2H] into reg (reuse)
  for (int idx = tid; idx < TT * 2 * HH; idx += 256) {
    int t = idx / (2 * HH);
    int c = idx % (2 * HH);
    float s = 0.f;
    if (c < HH) {
      for (int s0 = 0; s0 < TT; ++s0)
        s += (sAttn[t][s0 * 2] + sAttn[t][s0 * 2 + 1]) * sNf[s0][c];
    } else {
      int h = c - HH;
      float w0 = 0.f, w1 = 0.f;
      for (int s0 = 0; s0 < TT; ++s0) { w0 += sAttn[t][s0 * 2]; w1 += sAttn[t][s0 * 2 + 1]; }
      s = w0 * E[h] + w1 * E[HH + h];
    }
    reg[idx] = s;
  }
  __syncthreads();
  int hasEdges = 0;
  for (int t = 0; t < TT; ++t) hasEdges |= sHasIn[t];
  // (i) upd = relu(agg @ W_gat + b_gat); scatter-add at center rows
  if (hasEdges) {
    for (int idx = tid; idx < TT * HH; idx += 256) {
      int t = idx / HH, h = idx % HH;
      float nn;
      if (sHasIn[t]) {
        const float* rt = reg + t * 2 * HH;
        float acc = bg[h];
        for (int c = 0; c < 2 * HH; ++c) acc += rt[c] * Wg[c * HH + h];
        nn = fmaxf(acc, 0.f);
      } else {
        nn = sNf[t][h];
      }
      int ctr = (ast[t] + ost[t]) >> 1;
      atomicAdd(&embb[ctr * HH + h], nn);
    }
  }
}

// ---------------------------------------------------------------------------
// Kernel 3: build WcatT[16][256] = transpose of [Wc (256x6) | Wc1 (256x2) | 0]
// ---------------------------------------------------------------------------
__global__ void k_prepw(const float* __restrict__ Wc, const float* __restrict__ Wc1,
                        float* __restrict__ WcatT) {
  int k = threadIdx.x;  // 0..255
  for (int n = 0; n < 16; ++n) {
    float v;
    if (n < 6)      v = Wc[k * 6 + n];
    else if (n < 8) v = Wc1[k * 2 + (n - 6)];
    else            v = 0.f;
    WcatT[n * LL + k] = v;
  }
}

// ---------------------------------------------------------------------------
// Kernel 4: WMMA f32 GEMM  [1024,768] @ [768,512] -> A' (=A+bl1) and Bm
// One wave per 16x16 output tile, K-loop in steps of 4 (V_WMMA_F32_16X16X4_F32)
// ---------------------------------------------------------------------------
__global__ void __launch_bounds__(32) k_gemm1(
    const float* __restrict__ emb1,  // [1024,768]
    const float* __restrict__ W1,    // [1536,256]
    const float* __restrict__ bl1,   // [256]
    float* __restrict__ Ap,          // [1024,256]
    float* __restrict__ Bm)          // [1024,256]
{
  int m0 = blockIdx.x * 16;
  int n0 = blockIdx.y * 16;            // 0..511
  int lane = threadIdx.x;
  int half = lane >> 4, lrow = lane & 15;
  const float* arow = emb1 + (m0 + lrow) * HH;
  int nc = (n0 & 255) + lrow;          // column within 256-wide half
  const float* wbase = W1 + ((n0 >= 256) ? HH * LL : 0);

  v8f acc = {};
  for (int k = 0; k < HH; k += 4) {
    int ka = k + 2 * half;
    // A fragment (16x4 f32): lanes 0-15 carry K=ka,ka+1; lanes 16-31 K=ka(+2),ka+1(+3)
    v2f a = *(const v2f*)(arow + ka);
    v2f bf;
    bf.x = wbase[ka * LL + nc];
    bf.y = wbase[(ka + 1) * LL + nc];
    acc = __builtin_amdgcn_wmma_f32_16x16x4_f32(false, a, false, bf, (short)0, acc,
                                                false, false);
  }
  // D layout: VGPR r -> lanes 0-15: (M=r, N=lane); lanes 16-31: (M=8+r, N=lane-16)
  float* dst = (n0 < 256) ? Ap : Bm;
  float bias = (n0 < 256) ? bl1[nc] : 0.f;
#pragma unroll
  for (int r = 0; r < 8; ++r) {
    int row = m0 + half * 8 + r;
    dst[row * LL + nc] = acc[r] + bias;
  }
}

// ---------------------------------------------------------------------------
// Kernel 5: fused hidden build + LayerNorm + GELU + WMMA projection [256 -> 8]
// One wave handles 16 (i,j) pairs (fixed b,i; 16 consecutive j).
// ---------------------------------------------------------------------------
__global__ void __launch_bounds__(32) k_fused(
    const float* __restrict__ Ap,     // [B*L,256] includes bl1
    const float* __restrict__ Bm,     // [B*L,256]
    const float* __restrict__ WcatT,  // [16,256]
    const float* __restrict__ g1, const float* __restrict__ b1,  // [256]
    const float* __restrict__ bc, const float* __restrict__ bc1,
    const float* __restrict__ masks,  // [B,L,L]
    float* __restrict__ out)          // [B,L,L,8]
{
  int idx = blockIdx.x;
  int jt = idx & 15;
  int i  = (idx >> 4) & 255;
  int b  = idx >> 12;
  int lane = threadIdx.x;
  int half = lane >> 4, p = lane & 15;

  __shared__ float sAg[16][260];      // gelu'd activations, padded stride
  __shared__ float sW[16][256];       // WcatT tile

  for (int q = lane; q < 16 * 256; q += 32) sW[q >> 8][q & 255] = WcatT[q];

  const float* Ar = Ap + (b * LL + i) * LL;
  int j = jt * 16 + p;
  const float* Br = Bm + (b * LL + j) * LL;

  // pass 1: mean/var over k (two lanes per pair, each owns 128 k's)
  int kbase = half * 128;
  float s = 0.f, s2 = 0.f;
  for (int k = kbase; k < kbase + 128; ++k) {
    float h = Ar[k] + Br[k];
    s += h; s2 += h * h;
  }
  s  += __shfl_xor(s, 16, 32);
  s2 += __shfl_xor(s2, 16, 32);
  float mean = s * (1.f / 256.f);
  float var  = s2 * (1.f / 256.f) - mean * mean;
  float rstd = rsqrtf(var + 1e-5f);

  // pass 2: LN + exact GELU, stage to LDS
  for (int k = kbase; k < kbase + 128; ++k) {
    float h = Ar[k] + Br[k];
    float x = (h - mean) * rstd * g1[k] + b1[k];
    sAg[p][k] = 0.5f * x * (1.f + erff(x * 0.70710678118654752f));
  }
  __syncthreads();

  // WMMA projection: [16 pairs x 256] @ [256 x 16] via 64x V_WMMA_F32_16X16X4_F32
  v8f acc = {};
  for (int k = 0; k < 256; k += 4) {
    int ka = k + 2 * half;
    v2f a  = *(const v2f*)&sAg[p][ka];   // A: M=p, K=ka..ka+1
    v2f bf = *(const v2f*)&sW[p][ka];    // B: N=p, K=ka..ka+1
    acc = __builtin_amdgcn_wmma_f32_16x16x4_f32(false, a, false, bf, (short)0, acc,
                                                false, false);
  }

  // epilogue: bias, mask, write 8 channels
  int n = p;
  if (n < 8) {
    float bias = (n < 6) ? bc[n] : bc1[n - 6];
#pragma unroll
    for (int r = 0; r < 8; ++r) {
      int jo = jt * 16 + half * 8 + r;
      float mk = masks[(b * LL + i) * LL + jo];
      out[(((b * LL + i) * LL) + jo) * NCH + n] = (acc[r] + bias) * mk;
    }
  }
}

// ---------------------------------------------------------------------------
extern "C" void kernel_launch(void* const* d_in, const int* in_sizes, int n_in,
                              void* d_out, int out_size, void* d_ws, size_t ws_size,
                              hipStream_t stream) {
  const float* embeddings = (const float*)d_in[0];
  const float* masks      = (const float*)d_in[1];
  const int* asp_st  = (const int*)d_in[2];
  const int* asp_ed  = (const int*)d_in[3];
  const int* opi_st  = (const int*)d_in[4];
  const int* opi_ed  = (const int*)d_in[5];
  const int* sent_id = (const int*)d_in[6];
  const float* edge = (const float*)d_in[7];
  const float* Wtp  = (const float*)d_in[8];
  const float* btp  = (const float*)d_in[9];
  const float* Wa   = (const float*)d_in[10];
  const float* ba   = (const float*)d_in[11];
  const float* Wg   = (const float*)d_in[12];
  const float* bg   = (const float*)d_in[13];
  const float* g0   = (const float*)d_in[14];
  const float* be0  = (const float*)d_in[15];
  const float* W1   = (const float*)d_in[16];
  const float* bl1  = (const float*)d_in[17];
  const float* g1   = (const float*)d_in[18];
  const float* be1  = (const float*)d_in[19];
  const float* Wc   = (const float*)d_in[20];
  const float* bc   = (const float*)d_in[21];
  const float* Wc1  = (const float*)d_in[22];
  const float* bc1  = (const float*)d_in[23];

  float* ws    = (float*)d_ws;
  float* emb1  = ws + OFF_EMB1;
  float* Ap    = ws + OFF_APRIME;
  float* Bmw   = ws + OFF_BM;
  float* WcatT = ws + OFF_WCATT;
  float* trip  = ws + OFF_TRIP;
  float* out   = (float*)d_out;

  k_ln0<<<BB * LL, 256, 0, stream>>>(embeddings, g0, be0, emb1);
  k_triplet<<<BB, 256, 0, stream>>>(emb1, asp_st, asp_ed, opi_st, opi_ed, sent_id,
                                    edge, Wtp, btp, Wa, ba, Wg, bg, trip);
  k_prepw<<<1, 256, 0, stream>>>(Wc, Wc1, WcatT);
  dim3 g4(64, 32);
  k_gemm1<<<g4, 32, 0, stream>>>(emb1, W1, bl1, Ap, Bmw);
  k_fused<<<BB * LL * (LL / 16), 32, 0, stream>>>(Ap, Bmw, WcatT, g1, be1, bc, bc1,
                                                  masks, out);
}